// SampleNetDC3D_38122129719633
// MI455X (gfx1250) — compile-verified
//
#include <hip/hip_runtime.h>
#include <cfloat>

// ---------------------------------------------------------------------------
// Problem constants (match reference)
// ---------------------------------------------------------------------------
#define BGR   8        // graphs
#define PP    4096     // points per graph
#define NPT   (BGR*PP) // 32768 points
#define KNN   20       // neighbors
#define LL    10       // neighbors used for covariance
#define FF    32       // features
#define KS    5        // spline kernel size
#define NCLS  40
#define H1    256
#define YROWS 24       // (N*F*3)/(P*F)

typedef __attribute__((ext_vector_type(2))) float v2f;
typedef __attribute__((ext_vector_type(8))) float v8f;

// ---------------------------------------------------------------------------
// Kernel 0: zero accumulators
// ---------------------------------------------------------------------------
__global__ void init_kernel(float* __restrict__ stats, float* __restrict__ ysAcc) {
    int t = threadIdx.x;
    if (t < 64) stats[t] = 0.0f;
    for (int j = t; j < YROWS * FF; j += 256) ysAcc[j] = 0.0f;
}

// ---------------------------------------------------------------------------
// Kernel 1: fused KNN.
//  grid = 8 graphs * 64 row-groups, block = 128 threads (4 waves).
//  Whole graph (4096 float4 = pos + |p|^2) staged in LDS.
//  Each wave owns 16 query rows; loops over 256 column tiles of 16.
//  Gram tile via v_wmma_f32_16x16x4_f32; dist = qn + cn - 2*dot;
//  tile bounced through wave-private LDS, 2 lanes/row keep sorted top-20.
// ---------------------------------------------------------------------------
#define KNN_SMEM (PP*16 + 4*288*4 + 4*16*2*KNN*4 + 4*16*2*KNN*4)  // 90624 B

__global__ __launch_bounds__(128) void knn_kernel(const float* __restrict__ pos,
                                                  int* __restrict__ nbr) {
    extern __shared__ unsigned char smem[];
    float4* pts  = (float4*)smem;                               // 65536 B
    float*  tile = (float*)(smem + PP * 16);                    // 4*288 f
    float*  mrgD = (float*)(smem + PP * 16 + 4 * 288 * 4);      // 2560 f
    int*    mrgI = (int*)  (smem + PP * 16 + 4 * 288 * 4 + 10240);

    const int graph  = blockIdx.x >> 6;
    const int rowBlk = blockIdx.x & 63;
    const int tid  = threadIdx.x;
    const int wave = tid >> 5;
    const int lane = tid & 31;

    // stage whole graph into LDS with precomputed |p|^2
    for (int j = tid; j < PP; j += 128) {
        const float* pp = pos + (size_t)(graph * PP + j) * 3;
        float x = pp[0], y = pp[1], z = pp[2];
        pts[j] = make_float4(x, y, z, x * x + y * y + z * z);
    }
    __syncthreads();

    const int rowBase = rowBlk * 64 + wave * 16;   // within-graph query base
    const int half    = lane >> 4;                 // 0: lanes 0-15, 1: 16-31
    const int l16     = lane & 15;

    // A matrix 16x4 (rows = queries). Layout: lanes0-15 K=0/1, lanes16-31 K=2/3
    float4 q = pts[rowBase + l16];
    v2f A;
    A.x = half ? q.z : q.x;   // K=0 / K=2
    A.y = half ? 0.f : q.y;   // K=1 / K=3 (pad)

    // per-lane row norms for its 8 C-matrix rows (M = r + 8*half)
    float qn[8];
#pragma unroll
    for (int r = 0; r < 8; ++r) qn[r] = pts[rowBase + r + half * 8].w;

    const float INF = __builtin_inff();
    float dl[KNN]; int il[KNN];
#pragma unroll
    for (int t = 0; t < KNN; ++t) { dl[t] = INF; il[t] = 0; }

    const int selRow  = lane >> 1;        // selection mapping: 2 lanes per row
    const int selCol0 = (lane & 1) * 8;
    float* myTile = tile + wave * 288;    // 16 x 18 floats

    for (int ct = 0; ct < PP / 16; ++ct) {
        const int colb = ct * 16;
        float4 c = pts[colb + l16];
        v2f Bm;
        Bm.x = half ? c.z : c.x;
        Bm.y = half ? 0.f : c.y;

        v8f acc = {};
        acc = __builtin_amdgcn_wmma_f32_16x16x4_f32(
            false, A, false, Bm, (short)0, acc, false, false);

#pragma unroll
        for (int r = 0; r < 8; ++r) {
            const int rowM = r + half * 8;
            float dist = qn[r] + c.w - 2.0f * acc[r];
            if (colb + l16 == rowBase + rowM) dist = INF;   // exclude self
            myTile[rowM * 18 + l16] = dist;
        }
        asm volatile("s_wait_dscnt 0" ::: "memory");
        __builtin_amdgcn_wave_barrier();

#pragma unroll
        for (int j = 0; j < 8; ++j) {
            float d  = myTile[selRow * 18 + selCol0 + j];
            int  idx = colb + selCol0 + j;
            if (d < dl[KNN - 1]) {
                dl[KNN - 1] = d; il[KNN - 1] = idx;
#pragma unroll
                for (int t = KNN - 1; t > 0; --t) {
                    if (dl[t] < dl[t - 1]) {
                        float td = dl[t]; dl[t] = dl[t - 1]; dl[t - 1] = td;
                        int   ti = il[t]; il[t] = il[t - 1]; il[t - 1] = ti;
                    }
                }
            }
        }
        __builtin_amdgcn_wave_barrier();
        asm volatile("s_wait_dscnt 0" ::: "memory");
    }

    // merge the two per-row sorted lists
    {
        const int base = ((wave * 16 + selRow) * 2 + (lane & 1)) * KNN;
#pragma unroll
        for (int t = 0; t < KNN; ++t) { mrgD[base + t] = dl[t]; mrgI[base + t] = il[t]; }
    }
    asm volatile("s_wait_dscnt 0" ::: "memory");
    __builtin_amdgcn_wave_barrier();

    if ((lane & 1) == 0) {
        const int ba = (wave * 16 + selRow) * 2 * KNN;
        const float* da = mrgD + ba;           const float* db = da + KNN;
        const int*   ia = mrgI + ba;           const int*   ib = ia + KNN;
        int a = 0, b = 0;
        const int rowg = graph * PP + rowBase + selRow;
        int* out = nbr + (size_t)rowg * KNN;
        for (int t = 0; t < KNN; ++t) {
            if (da[a] <= db[b]) { out[t] = graph * PP + ia[a]; ++a; }
            else                { out[t] = graph * PP + ib[b]; ++b; }
        }
    }
}

// ---------------------------------------------------------------------------
// Kernel 2: per-point geometry (cov, power-iter eig3), spline conv, node
// features, batch-stat accumulation. Thread per point; Wsp staged in LDS.
// ---------------------------------------------------------------------------
__device__ __forceinline__ void power_iter(float m00, float m01, float m02,
                                           float m11, float m12, float m22,
                                           float& vx, float& vy, float& vz) {
    float x = 0.57735026919f, y = x, z = x;
#pragma unroll
    for (int it = 0; it < 5; ++it) {
        float nx = m00 * x + m01 * y + m02 * z;
        float ny = m01 * x + m11 * y + m12 * z;
        float nz = m02 * x + m12 * y + m22 * z;
        float nn = sqrtf(nx * nx + ny * ny + nz * nz) + 1e-12f;
        x = nx / nn; y = ny / nn; z = nz / nn;
    }
    vx = x; vy = y; vz = z;
}

__global__ __launch_bounds__(256) void geom_kernel(
    const float* __restrict__ pos,  const float* __restrict__ Wsp,
    const float* __restrict__ root, const float* __restrict__ bias,
    const int*   __restrict__ nbr,
    float* __restrict__ node, float* __restrict__ nrm,
    float* __restrict__ stats) {
    __shared__ float wsp[KS * KS * KS * FF];   // 16000 B
    __shared__ float bsum[FF];
    __shared__ float bsq[FF];
    const int tid = threadIdx.x;
    for (int j = tid; j < KS * KS * KS * FF; j += 256) wsp[j] = Wsp[j];
    if (tid < FF) { bsum[tid] = 0.f; bsq[tid] = 0.f; }
    __syncthreads();

    const int i = blockIdx.x * 256 + tid;
    const float px = pos[3 * i + 0], py = pos[3 * i + 1], pz = pos[3 * i + 2];
    const int* nb = nbr + (size_t)i * KNN;

    // covariance over first L neighbors
    float c00 = 0, c01 = 0, c02 = 0, c11 = 0, c12 = 0, c22 = 0;
    for (int k = 0; k < LL; ++k) {
        int n = nb[k];
        float dx = pos[3 * n + 0] - px, dy = pos[3 * n + 1] - py, dz = pos[3 * n + 2] - pz;
        c00 += dx * dx; c01 += dx * dy; c02 += dx * dz;
        c11 += dy * dy; c12 += dy * dz; c22 += dz * dz;
    }

    float v1x, v1y, v1z;
    power_iter(c00, c01, c02, c11, c12, c22, v1x, v1y, v1z);
    float mx = c00 * v1x + c01 * v1y + c02 * v1z;
    float my = c01 * v1x + c11 * v1y + c12 * v1z;
    float mz = c02 * v1x + c12 * v1y + c22 * v1z;
    float lam = v1x * mx + v1y * my + v1z * mz;

    float v2x, v2y, v2z;
    power_iter(c00 - lam * v1x * v1x, c01 - lam * v1x * v1y, c02 - lam * v1x * v1z,
               c11 - lam * v1y * v1y, c12 - lam * v1y * v1z, c22 - lam * v1z * v1z,
               v2x, v2y, v2z);

    float v3x = v1y * v2z - v1z * v2y;
    float v3y = v1z * v2x - v1x * v2z;
    float v3z = v1x * v2y - v1y * v2x;
    float n3 = sqrtf(v3x * v3x + v3y * v3y + v3z * v3z) + 1e-12f;
    v3x /= n3; v3y /= n3; v3z /= n3;

    // pass 1: sign and max-abs of dirc over all K neighbors
    float sum2 = 0.f, maxabs = 0.f;
    for (int k = 0; k < KNN; ++k) {
        int n = nb[k];
        float dx = pos[3 * n + 0] - px, dy = pos[3 * n + 1] - py, dz = pos[3 * n + 2] - pz;
        float d0 = dx * v1x + dy * v1y + dz * v1z;
        float d1 = dx * v2x + dy * v2y + dz * v2z;
        float d2 = dx * v3x + dy * v3y + dz * v3z;
        sum2 += d2;
        maxabs = fmaxf(maxabs, fmaxf(fabsf(d0), fmaxf(fabsf(d1), fabsf(d2))));
    }
    const float sgn   = (sum2 > 0.f) ? 1.f : ((sum2 < 0.f) ? -1.f : 0.f);
    const float scale = 0.5f / maxabs;

    // pass 2: spline conv message (degree 1, open splines, KS=5, dim=3)
    float msg[FF];
#pragma unroll
    for (int f = 0; f < FF; ++f) msg[f] = 0.f;

    for (int k = 0; k < KNN; ++k) {
        int n = nb[k];
        float dx = pos[3 * n + 0] - px, dy = pos[3 * n + 1] - py, dz = pos[3 * n + 2] - pz;
        float p0 = (dx * v1x + dy * v1y + dz * v1z) * scale + 0.5f;
        float p1 = (dx * v2x + dy * v2y + dz * v2z) * scale + 0.5f;
        float p2 = (dx * v3x + dy * v3y + dz * v3z) * sgn * scale + 0.5f;
        float u0 = p0 * (KS - 1), u1 = p1 * (KS - 1), u2 = p2 * (KS - 1);
        float f0 = floorf(u0), f1 = floorf(u1), f2 = floorf(u2);
        float r0 = u0 - f0, r1 = u1 - f1, r2 = u2 - f2;
        int i0 = (int)f0, i1 = (int)f1, i2 = (int)f2;
#pragma unroll
        for (int s = 0; s < 8; ++s) {
            const int b0 = (s >> 2) & 1, b1 = (s >> 1) & 1, b2 = s & 1;
            float w = (b0 ? r0 : 1.f - r0) * (b1 ? r1 : 1.f - r1) * (b2 ? r2 : 1.f - r2);
            int ix = max(0, min(i0 + b0, KS - 1));
            int iy = max(0, min(i1 + b1, KS - 1));
            int iz = max(0, min(i2 + b2, KS - 1));
            const float* wr = &wsp[((ix * KS + iy) * KS + iz) * FF];
#pragma unroll
            for (int f = 0; f < FF; ++f) msg[f] += w * wr[f];
        }
    }

    const float invK = 1.0f / (float)KNN;
#pragma unroll
    for (int f = 0; f < FF; ++f) {
        float nf = msg[f] * invK + root[f] + bias[f];
        node[(size_t)i * FF + f] = nf;
        atomicAdd(&bsum[f], nf);
        atomicAdd(&bsq[f], nf * nf);
    }
    nrm[3 * i + 0] = v3x; nrm[3 * i + 1] = v3y; nrm[3 * i + 2] = v3z;

    __syncthreads();
    if (tid < FF) {
        atomicAdd(&stats[tid], bsum[tid]);
        atomicAdd(&stats[FF + tid], bsq[tid]);
    }
}

// ---------------------------------------------------------------------------
// Kernel 3: batchnorm + sigmoid(out3) + reshape(24,4096,32).mean(1) accum
// ---------------------------------------------------------------------------
__global__ __launch_bounds__(256) void tail_kernel(
    const float* __restrict__ node, const float* __restrict__ nrm,
    const float* __restrict__ stats,
    const float* __restrict__ gamma, const float* __restrict__ beta,
    float* __restrict__ ysAcc) {
    __shared__ float ysl[YROWS * FF];
    const int tid = threadIdx.x;
    for (int j = tid; j < YROWS * FF; j += 256) ysl[j] = 0.f;
    __syncthreads();

    const int i = blockIdx.x * 256 + tid;
    const float n0 = nrm[3 * i], n1 = nrm[3 * i + 1], n2 = nrm[3 * i + 2];
    const float invN = 1.0f / (float)NPT;

    for (int f = 0; f < FF; ++f) {
        float mu  = stats[f] * invN;
        float var = stats[FF + f] * invN - mu * mu;
        float xb  = gamma[f] * (node[(size_t)i * FF + f] - mu) / sqrtf(var + 1e-5f) + beta[f];
        const int flatBase = i * (FF * 3) + f * 3;
        const float nc[3] = { n0, n1, n2 };
#pragma unroll
        for (int c = 0; c < 3; ++c) {
            float y = 1.0f / (1.0f + expf(-xb * nc[c]));
            int flat = flatBase + c;
            int row = flat >> 17;       // / (4096*32)
            int col = flat & 31;        // % 32
            atomicAdd(&ysl[row * FF + col], y);
        }
    }
    __syncthreads();
    for (int j = tid; j < YROWS * FF; j += 256) atomicAdd(&ysAcc[j], ysl[j]);
}

// ---------------------------------------------------------------------------
// Kernel 4: tiny MLP (24x32 @ 32x256, elu, @ 256x40, log_softmax), one block
// ---------------------------------------------------------------------------
__global__ __launch_bounds__(256) void mlp_kernel(
    const float* __restrict__ ysAcc,
    const float* __restrict__ W1, const float* __restrict__ b1,
    const float* __restrict__ W2, const float* __restrict__ b2,
    float* __restrict__ out) {
    __shared__ float ys[YROWS * FF];
    __shared__ float h[YROWS * H1];
    __shared__ float lg[YROWS * NCLS];
    const int tid = threadIdx.x;
    for (int j = tid; j < YROWS * FF; j += 256) ys[j] = ysAcc[j] * (1.0f / (float)PP);
    __syncthreads();

    // hidden: thread tid owns column tid
    {
        const float bb = b1[tid];
        for (int r = 0; r < YROWS; ++r) {
            float s = bb;
#pragma unroll
            for (int k = 0; k < FF; ++k) s += ys[r * FF + k] * W1[k * H1 + tid];
            h[r * H1 + tid] = (s > 0.f) ? s : (expm1f(s));
        }
    }
    __syncthreads();

    for (int e = tid; e < YROWS * NCLS; e += 256) {
        int r = e / NCLS, j = e % NCLS;
        float s = b2[j];
        for (int k = 0; k < H1; ++k) s += h[r * H1 + k] * W2[k * NCLS + j];
        lg[e] = s;
    }
    __syncthreads();

    if (tid < YROWS) {
        float m = -FLT_MAX;
        for (int j = 0; j < NCLS; ++j) m = fmaxf(m, lg[tid * NCLS + j]);
        float se = 0.f;
        for (int j = 0; j < NCLS; ++j) se += expf(lg[tid * NCLS + j] - m);
        float ls = logf(se);
        for (int j = 0; j < NCLS; ++j)
            out[tid * NCLS + j] = lg[tid * NCLS + j] - m - ls;
    }
}

// ---------------------------------------------------------------------------
// launch
// ---------------------------------------------------------------------------
extern "C" void kernel_launch(void* const* d_in, const int* in_sizes, int n_in,
                              void* d_out, int out_size, void* d_ws, size_t ws_size,
                              hipStream_t stream) {
    const float* pos   = (const float*)d_in[0];
    const float* Wsp   = (const float*)d_in[1];
    const float* root  = (const float*)d_in[2];
    const float* bias  = (const float*)d_in[3];
    const float* gamma = (const float*)d_in[4];
    const float* beta  = (const float*)d_in[5];
    const float* W1    = (const float*)d_in[6];
    const float* b1    = (const float*)d_in[7];
    const float* W2    = (const float*)d_in[8];
    const float* b2    = (const float*)d_in[9];
    float* out = (float*)d_out;

    char* ws = (char*)d_ws;
    int*   nbr   = (int*)  (ws + 0);                         // 32768*20*4  = 2621440
    float* node  = (float*)(ws + 2621440);                   // 32768*32*4  = 4194304
    float* nrm   = (float*)(ws + 6815744);                   // 32768*3*4   = 393216
    float* stats = (float*)(ws + 7208960);                   // 64*4
    float* ysAcc = (float*)(ws + 7209216);                   // 768*4

    init_kernel<<<1, 256, 0, stream>>>(stats, ysAcc);
    knn_kernel<<<dim3(BGR * 64), dim3(128), KNN_SMEM, stream>>>(pos, nbr);
    geom_kernel<<<dim3(NPT / 256), dim3(256), 0, stream>>>(pos, Wsp, root, bias,
                                                           nbr, node, nrm, stats);
    tail_kernel<<<dim3(NPT / 256), dim3(256), 0, stream>>>(node, nrm, stats,
                                                           gamma, beta, ysAcc);
    mlp_kernel<<<1, 256, 0, stream>>>(ysAcc, W1, b1, W2, b2, out);
}